// RoIHeads_5909874999669
// MI455X (gfx1250) — compile-verified
//
#include <hip/hip_runtime.h>

typedef __attribute__((ext_vector_type(2))) float v2f;
typedef __attribute__((ext_vector_type(8))) float v8f;

#ifndef __has_builtin
#define __has_builtin(x) 0
#endif
#if __has_builtin(__builtin_amdgcn_wmma_f32_16x16x4_f32)
#define USE_WMMA_F32X4 1
#else
#define USE_WMMA_F32X4 0
#endif

namespace {
constexpr int   kB       = 16;
constexpr int   kN       = 4096;
constexpr int   kC       = 91;
constexpr int   kRows    = kB * kN;      // 65536 proposal rows
constexpr int   kCap     = 77824;        // 19*4096: provable max valid/image
constexpr float kScoreTh = 0.05f;
constexpr float kNmsTh   = 0.5f;
constexpr float kNeg     = -1e9f;
constexpr float kBoxClip = 4.1351665567423560f;  // log(1000/16)
constexpr int   kMaxDet  = 100;

// workspace byte offsets
constexpr size_t kOffCounters = 0;
constexpr size_t kOffRowMax   = 1024;
constexpr size_t kOffDenom    = kOffRowMax + (size_t)kRows * 4;            // 263168
constexpr size_t kOffCandB    = kOffDenom + (size_t)kRows * 4;             // 525312 (16B aligned)
constexpr size_t kOffCandS    = kOffCandB + (size_t)kB * kCap * 16;        // 20448256
constexpr size_t kOffCandL    = kOffCandS + (size_t)kB * kCap * 4;         // 25428992
} // namespace

__global__ void zero_counters(int* counters) {
    if (threadIdx.x < kB) counters[threadIdx.x] = 0;
}

// ---------------------------------------------------------------------------
// Kernel 1: per-row (proposal) softmax max + denominator.
// One wave32 handles 16 rows, matching the V_WMMA_F32_16X16X4_F32 A layout:
//   lane l (l<16):  row = l,     holds K = k0+0, k0+1
//   lane l (l>=16): row = l-16,  holds K = k0+2, k0+3
// B = 4x16 ones  =>  D[m][n] = sum_k exp(logit[m][k]) in every column.
// ---------------------------------------------------------------------------
__global__ __launch_bounds__(256) void softmax_denom_wmma(
    const float* __restrict__ logits,
    float* __restrict__ rowmax_out,
    float* __restrict__ denom_out) {
    const int wavesPerBlock = blockDim.x >> 5;
    const int gw   = blockIdx.x * wavesPerBlock + (threadIdx.x >> 5);
    const int lane = threadIdx.x & 31;
    const int rowBase = gw * 16;
    const int r = rowBase + (lane & 15);
    const int p = (lane >> 4) * 2;                 // 0 or 2: K parity group
    const float* __restrict__ lg = logits + (size_t)r * kC;

    // pass 1: stable-softmax row max (each lane sees half the row; xor-16 merge)
    float m = -3.0e38f;
    #pragma unroll
    for (int k0 = 0; k0 < 92; k0 += 4) {
        const int c0 = k0 + p;                     // <= 90 always
        const float a = lg[c0];
        const float b = (c0 + 1 < kC) ? lg[c0 + 1] : -3.0e38f;
        m = fmaxf(m, fmaxf(a, b));
    }
    m = fmaxf(m, __shfl_xor(m, 16, 32));

#if USE_WMMA_F32X4
    v8f acc = {0.f, 0.f, 0.f, 0.f, 0.f, 0.f, 0.f, 0.f};
    v2f ones; ones[0] = 1.0f; ones[1] = 1.0f;
    #pragma unroll
    for (int k0 = 0; k0 < 92; k0 += 4) {
        const int c0 = k0 + p;
        v2f a;
        a[0] = expf(lg[c0] - m);
        a[1] = (c0 + 1 < kC) ? expf(lg[c0 + 1] - m) : 0.0f;
        // (neg_a, A, neg_b, B, c_mod, C, reuse_a, reuse_b)
        acc = __builtin_amdgcn_wmma_f32_16x16x4_f32(false, a, false, ones,
                                                    (short)0, acc, false, false);
    }
    // D layout: VGPR r, lanes 0-15 -> M=r, lanes 16-31 -> M=8+r (sum replicated per column)
    if (lane == 0 || lane == 16) {
        const int rb = rowBase + ((lane >> 4) << 3);
        denom_out[rb + 0] = acc[0]; denom_out[rb + 1] = acc[1];
        denom_out[rb + 2] = acc[2]; denom_out[rb + 3] = acc[3];
        denom_out[rb + 4] = acc[4]; denom_out[rb + 5] = acc[5];
        denom_out[rb + 6] = acc[6]; denom_out[rb + 7] = acc[7];
    }
#else
    float s = 0.0f;
    #pragma unroll
    for (int k0 = 0; k0 < 92; k0 += 4) {
        const int c0 = k0 + p;
        s += expf(lg[c0] - m);
        if (c0 + 1 < kC) s += expf(lg[c0 + 1] - m);
    }
    s += __shfl_xor(s, 16, 32);
    if (lane < 16) denom_out[rowBase + lane] = s;
#endif
    if (lane < 16) rowmax_out[rowBase + lane] = m;
}

// ---------------------------------------------------------------------------
// Kernel 2: decode + clip + score + filter + wave-aggregated compaction.
// One wave per proposal row; lane handles classes c = 1+lane, +32, +64.
// ---------------------------------------------------------------------------
__global__ __launch_bounds__(256) void decode_compact(
    const float* __restrict__ logits,
    const float* __restrict__ boxreg,
    const float* __restrict__ proposals,
    const int*   __restrict__ image_hw,
    const float* __restrict__ rowmax,
    const float* __restrict__ denom,
    int*    __restrict__ counters,
    float4* __restrict__ candB,
    float*  __restrict__ candS,
    int*    __restrict__ candL) {
    const int gw   = blockIdx.x * (blockDim.x >> 5) + (threadIdx.x >> 5); // proposal row
    const int lane = threadIdx.x & 31;
    if (gw >= kRows) return;
    const int img = gw >> 12;                       // / kN

    const float px1 = proposals[gw * 4 + 0];
    const float py1 = proposals[gw * 4 + 1];
    const float px2 = proposals[gw * 4 + 2];
    const float py2 = proposals[gw * 4 + 3];
    const float w  = px2 - px1, h = py2 - py1;
    const float cx = px1 + 0.5f * w, cy = py1 + 0.5f * h;
    const float m  = rowmax[gw];
    const float dn = denom[gw];
    const float Hf = (float)image_hw[img * 2 + 0];
    const float Wf = (float)image_hw[img * 2 + 1];

    const float*  __restrict__ lg  = logits + (size_t)gw * kC;
    const float4* __restrict__ reg = reinterpret_cast<const float4*>(boxreg) + (size_t)gw * kC;
    __builtin_prefetch(reg, 0, 3);                  // global_prefetch_b8

    #pragma unroll
    for (int t = 0; t < 3; ++t) {
        const int c = 1 + lane + 32 * t;
        bool  valid = false;
        float score = 0.0f, x1 = 0.f, y1 = 0.f, x2 = 0.f, y2 = 0.f;
        if (c < kC) {
            score = expf(lg[c] - m) / dn;
            const float4 rg = reg[c];               // 16B aligned (row stride 364 floats)
            const float dx = rg.x * 0.1f;
            const float dy = rg.y * 0.1f;
            const float dw = fminf(rg.z * 0.2f, kBoxClip);
            const float dh = fminf(rg.w * 0.2f, kBoxClip);
            const float pcx = dx * w + cx;
            const float pcy = dy * h + cy;
            const float pw  = expf(dw) * w;
            const float ph  = expf(dh) * h;
            x1 = fminf(fmaxf(pcx - 0.5f * pw, 0.0f), Wf);
            y1 = fminf(fmaxf(pcy - 0.5f * ph, 0.0f), Hf);
            x2 = fminf(fmaxf(pcx + 0.5f * pw, 0.0f), Wf);
            y2 = fminf(fmaxf(pcy + 0.5f * ph, 0.0f), Hf);
            valid = (score > kScoreTh) && ((x2 - x1) >= 1.0f) && ((y2 - y1) >= 1.0f);
        }
        const unsigned mask = (unsigned)__ballot(valid);   // wave32: low 32 bits
        if (mask) {
            const int leader = __ffs(mask) - 1;
            int base = 0;
            if (lane == leader) base = atomicAdd(&counters[img], __popc(mask));
            base = __shfl(base, leader, 32);
            if (valid) {
                const int slot = base + __popc(mask & ((1u << lane) - 1u));
                if (slot < kCap) {
                    const size_t q = (size_t)img * kCap + slot;
                    candB[q] = make_float4(x1, y1, x2, y2);
                    candS[q] = score;
                    candL[q] = c;
                }
            }
        }
    }
}

// ---------------------------------------------------------------------------
// Kernel 3: greedy class-aware NMS, one block per image (candidates are L2
// resident).  Cross-label IoU is exactly 0 under the reference's per-label
// offset, so we only suppress same-label boxes on raw coordinates.
// ---------------------------------------------------------------------------
__global__ __launch_bounds__(1024) void nms_kernel(
    const int*    __restrict__ counters,
    const float4* __restrict__ candB,
    float*        __restrict__ candS,     // mutated: acts as the `ms` array
    const int*    __restrict__ candL,
    float*        __restrict__ out) {
    const int img = blockIdx.x;
    const int tid = threadIdx.x;
    int cnt = counters[img];
    if (cnt > kCap) cnt = kCap;

    const float4* bx = candB + (size_t)img * kCap;
    float*        ms = candS + (size_t)img * kCap;
    const int*    lb = candL + (size_t)img * kCap;

    float* obox = out + (size_t)img * kMaxDet * 4;
    float* oscr = out + (size_t)kB * kMaxDet * 4 + (size_t)img * kMaxDet;
    float* olab = out + (size_t)kB * kMaxDet * 5 + (size_t)img * kMaxDet;

    // outputs are poisoned by the harness: zero our region first
    for (int i = tid; i < kMaxDet * 6; i += 1024) {
        if (i < 400)      obox[i] = 0.0f;
        else if (i < 500) oscr[i - 400] = 0.0f;
        else              olab[i - 500] = 0.0f;
    }
    __syncthreads();

    __shared__ float  ss[1024];
    __shared__ int    si[1024];
    __shared__ float4 sBox;
    __shared__ int    sLab;
    __shared__ int    sDone;

    for (int it = 0; it < kMaxDet; ++it) {
        // ---- argmax (first-index tie-break) ----
        float bs = -3.0e38f;
        int   bi = 0x7FFFFFFF;
        for (int i = tid; i < cnt; i += 1024) {
            const float s = ms[i];
            if (s > bs || (s == bs && i < bi)) { bs = s; bi = i; }
        }
        ss[tid] = bs; si[tid] = bi;
        __syncthreads();
        #pragma unroll
        for (int off = 512; off > 0; off >>= 1) {
            if (tid < off) {
                const float s2 = ss[tid + off];
                const int   i2 = si[tid + off];
                if (s2 > ss[tid] || (s2 == ss[tid] && i2 < si[tid])) {
                    ss[tid] = s2; si[tid] = i2;
                }
            }
            __syncthreads();
        }
        if (tid == 0) {
            const float s = ss[0];
            const int   b = si[0];
            if (s > 0.0f) {
                const float4 bb = bx[b];
                sBox = bb; sLab = lb[b]; sDone = 0;
                obox[it * 4 + 0] = bb.x; obox[it * 4 + 1] = bb.y;
                obox[it * 4 + 2] = bb.z; obox[it * 4 + 3] = bb.w;
                oscr[it] = s;
                olab[it] = (float)lb[b];
                ms[b] = kNeg;
            } else {
                sDone = 1;                 // all remaining outputs stay zero
            }
        }
        __syncthreads();
        if (sDone) break;

        // ---- suppress same-label boxes with IoU > 0.5 ----
        const float4 q  = sBox;
        const int    ql = sLab;
        const float  a1 = (q.z - q.x) * (q.w - q.y);
        for (int i = tid; i < cnt; i += 1024) {
            if (lb[i] == ql) {
                const float4 o = bx[i];
                const float lx = fmaxf(q.x, o.x), ly = fmaxf(q.y, o.y);
                const float rx = fminf(q.z, o.z), ry = fminf(q.w, o.w);
                const float iw = fmaxf(rx - lx, 0.0f), ih = fmaxf(ry - ly, 0.0f);
                const float inter = iw * ih;
                const float a2 = (o.z - o.x) * (o.w - o.y);
                const float iou = inter / (a1 + a2 - inter);
                if (iou > kNmsTh) ms[i] = kNeg;
            }
        }
        __syncthreads();
    }
}

extern "C" void kernel_launch(void* const* d_in, const int* in_sizes, int n_in,
                              void* d_out, int out_size, void* d_ws, size_t ws_size,
                              hipStream_t stream) {
    (void)in_sizes; (void)n_in; (void)out_size; (void)ws_size;
    const float* logits    = (const float*)d_in[0];   // [B*N, 91]
    const float* boxreg    = (const float*)d_in[1];   // [B*N, 364]
    const float* proposals = (const float*)d_in[2];   // [B, N, 4]
    const int*   image_hw  = (const int*)  d_in[3];   // [B, 2]

    char* ws = (char*)d_ws;
    int*    counters = (int*)   (ws + kOffCounters);
    float*  rowmax   = (float*) (ws + kOffRowMax);
    float*  denom    = (float*) (ws + kOffDenom);
    float4* candB    = (float4*)(ws + kOffCandB);
    float*  candS    = (float*) (ws + kOffCandS);
    int*    candL    = (int*)   (ws + kOffCandL);
    float*  out      = (float*) d_out;

    zero_counters<<<1, 32, 0, stream>>>(counters);
    // 65536 rows / 16 rows-per-wave / 8 waves-per-block = 512 blocks
    softmax_denom_wmma<<<512, 256, 0, stream>>>(logits, rowmax, denom);
    // 65536 proposals / 8 waves-per-block = 8192 blocks
    decode_compact<<<8192, 256, 0, stream>>>(logits, boxreg, proposals, image_hw,
                                             rowmax, denom, counters,
                                             candB, candS, candL);
    nms_kernel<<<kB, 1024, 0, stream>>>(counters, candB, candS, candL, out);
}